// DynamicRNNDecoder_16527034155311
// MI455X (gfx1250) — compile-verified
//
#include <hip/hip_runtime.h>
#include <math.h>

// ---------------------------------------------------------------------------
// Dims (match reference)
#define BATCH 128
#define ZDIM  128
#define RDIM  256
#define FDIM  128
#define HIDD  128
#define AGGD  64
#define MAXN  64
#define G3    768   // 3*R

typedef float v2f __attribute__((ext_vector_type(2)));
typedef float v8f __attribute__((ext_vector_type(8)));

// ---------------------------------------------------------------------------
// Generic fp32 WMMA GEMM: one wave (32 thr) per 16x16 output tile.
//   C[M,N] = A[M,K] @ B[K,N]  (+ bias[n] if mode&1)  (relu if mode&2)
// grid = (M/16, N/16), block = 32.  K must be a multiple of 4.
// CDNA5 V_WMMA_F32_16X16X4_F32 fragment layouts (ISA 7.12.2):
//   A 16x4 : lane m = lane&15 ; k = 2*(lane>>4) + vgpr
//   B 4x16 : lane n = lane&15 ; k = 2*(lane>>4) + vgpr
//   C 16x16: row = r + 8*(lane>>4), col = lane&15
__global__ void k_gemm(const float* __restrict__ A, const float* __restrict__ Bm,
                       const float* __restrict__ bias, float* __restrict__ C,
                       int K, int N, int mode) {
  const int lane = threadIdx.x;
  const int l15  = lane & 15;
  const int half = lane >> 4;
  const int m0   = blockIdx.x * 16;
  const int n0   = blockIdx.y * 16;
  const float* arow = A + (size_t)(m0 + l15) * K;
  const int ncol = n0 + l15;

  v8f acc = {};
  for (int k0 = 0; k0 < K; k0 += 4) {
    const int ka = k0 + 2 * half;
    v2f a, b;
    a.x = arow[ka];
    a.y = arow[ka + 1];
    b.x = Bm[(size_t)ka * N + ncol];
    b.y = Bm[(size_t)(ka + 1) * N + ncol];
    acc = __builtin_amdgcn_wmma_f32_16x16x4_f32(false, a, false, b,
                                                (short)0, acc, false, false);
  }
  const float bv = (mode & 1) ? bias[ncol] : 0.0f;
#pragma unroll
  for (int r = 0; r < 8; ++r) {
    float v = acc[r] + bv;
    if (mode & 2) v = fmaxf(v, 0.0f);
    C[(size_t)(m0 + r + 8 * half) * N + ncol] = v;
  }
}

// DeepSet incremental update: S[m,n] += relu(acc + bg1[n]) - relu(bg1[n])
__global__ void k_gemm_supd(const float* __restrict__ A, const float* __restrict__ Bm,
                            const float* __restrict__ bias, float* __restrict__ S,
                            int K, int N) {
  const int lane = threadIdx.x;
  const int l15  = lane & 15;
  const int half = lane >> 4;
  const int m0   = blockIdx.x * 16;
  const int n0   = blockIdx.y * 16;
  const float* arow = A + (size_t)(m0 + l15) * K;
  const int ncol = n0 + l15;

  v8f acc = {};
  for (int k0 = 0; k0 < K; k0 += 4) {
    const int ka = k0 + 2 * half;
    v2f a, b;
    a.x = arow[ka];
    a.y = arow[ka + 1];
    b.x = Bm[(size_t)ka * N + ncol];
    b.y = Bm[(size_t)(ka + 1) * N + ncol];
    acc = __builtin_amdgcn_wmma_f32_16x16x4_f32(false, a, false, b,
                                                (short)0, acc, false, false);
  }
  const float bv = bias[ncol];
  const float rb = fmaxf(bv, 0.0f);
#pragma unroll
  for (int r = 0; r < 8; ++r) {
    const size_t idx = (size_t)(m0 + r + 8 * half) * N + ncol;
    S[idx] += fmaxf(acc[r] + bv, 0.0f) - rb;
  }
}

// Feature-module 2nd GEMM, storing logits transposed into d_out [B, MAXN, F]:
//   row m of A enumerates (node = m>>7, batch = m&127)
__global__ void k_gemm_fout(const float* __restrict__ A, const float* __restrict__ Bm,
                            const float* __restrict__ bias, float* __restrict__ outF,
                            int K, int N) {
  const int lane = threadIdx.x;
  const int l15  = lane & 15;
  const int half = lane >> 4;
  const int m0   = blockIdx.x * 16;
  const int n0   = blockIdx.y * 16;
  const float* arow = A + (size_t)(m0 + l15) * K;
  const int ncol = n0 + l15;

  v8f acc = {};
  for (int k0 = 0; k0 < K; k0 += 4) {
    const int ka = k0 + 2 * half;
    v2f a, b;
    a.x = arow[ka];
    a.y = arow[ka + 1];
    b.x = Bm[(size_t)ka * N + ncol];
    b.y = Bm[(size_t)(ka + 1) * N + ncol];
    acc = __builtin_amdgcn_wmma_f32_16x16x4_f32(false, a, false, b,
                                                (short)0, acc, false, false);
  }
  const float bv = bias[ncol];
#pragma unroll
  for (int r = 0; r < 8; ++r) {
    const int m = m0 + r + 8 * half;
    const int node = m >> 7;
    const int b    = m & 127;
    outF[(size_t)b * (MAXN * FDIM) + node * FDIM + ncol] = acc[r] + bv;
  }
}

// ---------------------------------------------------------------------------
// Prologue zero/init: h_nodes[1..], P[1..], hc, S=64*relu(bg1), e row 0
__global__ void k_zero(float* __restrict__ h_nodes, float* __restrict__ P,
                       float* __restrict__ hc, float* __restrict__ S,
                       const float* __restrict__ bg1, float* __restrict__ e_out) {
  const size_t idx = (size_t)blockIdx.x * blockDim.x + threadIdx.x;
  if (idx < (size_t)(MAXN - 1) * BATCH * RDIM) h_nodes[(size_t)BATCH * RDIM + idx] = 0.0f;
  if (idx < (size_t)(MAXN - 1) * BATCH * HIDD) P[(size_t)BATCH * HIDD + idx] = 0.0f;
  if (idx < (size_t)MAXN * BATCH) hc[idx] = 0.0f;
  if (idx < (size_t)BATCH * AGGD) {
    const int a = (int)(idx & (AGGD - 1));
    S[idx] = 64.0f * fmaxf(bg1[a], 0.0f);
  }
  if (idx < (size_t)BATCH * MAXN) {
    const int b = (int)(idx >> 6), j = (int)(idx & 63);
    e_out[(size_t)b * (MAXN * MAXN) + j] = 0.0f;  // e row i=0
  }
}

// Edge MLP 2nd layer + sigmoid + mask, updates hc and e output.
// grid = MAXN (node j), block = BATCH (batch b)
__global__ void k_edge(const float* __restrict__ Q, const float* __restrict__ P,
                       const float* __restrict__ be1, const float* __restrict__ we2,
                       const float* __restrict__ be2, float* __restrict__ erow,
                       float* __restrict__ hc, float* __restrict__ e_out, int i) {
  const int j = blockIdx.x;
  const int b = threadIdx.x;
  const float* q = Q + (size_t)b * HIDD;
  const float* p = P + ((size_t)j * BATCH + b) * HIDD;
  float acc = 0.0f;
  for (int h = 0; h < HIDD; ++h) {
    const float v = fmaxf(q[h] + p[h] + be1[h], 0.0f);
    acc += v * we2[h];
  }
  const float pre = acc + be2[0];
  const float er = (j < i) ? 1.0f / (1.0f + expf(-pre)) : 0.0f;
  erow[j * BATCH + b] = er;
  hc[j * BATCH + b] += er;
  e_out[(size_t)b * (MAXN * MAXN) + i * MAXN + j] = er;
}

// Masked ancestor/sink sums. grid = BATCH, block = RDIM
__global__ void k_aggr(const float* __restrict__ erow, const float* __restrict__ hc,
                       const float* __restrict__ h, float* __restrict__ anc,
                       float* __restrict__ snk) {
  const int b = blockIdx.x;
  const int r = threadIdx.x;
  float a = 0.0f, s = 0.0f;
  for (int j = 0; j < MAXN; ++j) {
    const float hv = h[((size_t)j * BATCH + b) * RDIM + r];
    if (erow[j * BATCH + b] >= 0.5f) a += hv;
    if (hc[j * BATCH + b] < 0.5f)    s += hv;
  }
  anc[(size_t)b * RDIM + r] = a;
  snk[(size_t)b * RDIM + r] = s;
}

// GRU gate nonlinearity (reset_after, [z,r,h]). grid = BATCH, block = RDIM
__global__ void k_gru(const float* __restrict__ XZ, const float* __restrict__ HZ,
                      const float* __restrict__ anc, float* __restrict__ h_i) {
  const int b = blockIdx.x;
  const int r = threadIdx.x;
  const float xz = XZ[(size_t)b * G3 + r];
  const float xr = XZ[(size_t)b * G3 + RDIM + r];
  const float xh = XZ[(size_t)b * G3 + 2 * RDIM + r];
  const float hz = HZ[(size_t)b * G3 + r];
  const float hr = HZ[(size_t)b * G3 + RDIM + r];
  const float hhl = HZ[(size_t)b * G3 + 2 * RDIM + r];
  const float z  = 1.0f / (1.0f + expf(-(xz + hz)));
  const float rr = 1.0f / (1.0f + expf(-(xr + hr)));
  const float hh = tanhf(xh + rr * hhl);
  const float st = anc[(size_t)b * RDIM + r];
  h_i[(size_t)b * RDIM + r] = z * st + (1.0f - z) * hh;
}

// lambdas 2nd layer + N output. 1 block, BATCH threads.
__global__ void k_lambda(const float* __restrict__ lamh, const float* __restrict__ wn2,
                         const float* __restrict__ bn2, float* __restrict__ outLam,
                         int* __restrict__ outN) {
  const int b = threadIdx.x;
  float acc = 0.0f;
  for (int h = 0; h < HIDD; ++h) acc += lamh[(size_t)b * HIDD + h] * wn2[h];
  const float lam = fmaxf(acc + bn2[0], 0.0f);
  outLam[b] = lam;
  int n = (int)ceilf(lam);
  if (n > MAXN) n = MAXN;
  outN[b] = n;
}

// In-place softmax over 128-wide rows of the f output. grid = MAXN*BATCH, block = 128
__global__ void k_softmax(float* __restrict__ fout) {
  __shared__ float red[128];
  const int row = blockIdx.x;
  const int t = threadIdx.x;
  float* rp = fout + (size_t)row * FDIM;
  const float v = rp[t];
  red[t] = v;
  __syncthreads();
  for (int s = 64; s > 0; s >>= 1) {
    if (t < s) red[t] = fmaxf(red[t], red[t + s]);
    __syncthreads();
  }
  const float mx = red[0];
  __syncthreads();
  const float e = expf(v - mx);
  red[t] = e;
  __syncthreads();
  for (int s = 64; s > 0; s >>= 1) {
    if (t < s) red[t] += red[t + s];
    __syncthreads();
  }
  rp[t] = e / red[0];
}

// ---------------------------------------------------------------------------
extern "C" void kernel_launch(void* const* d_in, const int* in_sizes, int n_in,
                              void* d_out, int out_size, void* d_ws, size_t ws_size,
                              hipStream_t stream) {
  (void)in_sizes; (void)n_in; (void)out_size; (void)ws_size;
  const float* x   = (const float*)d_in[0];
  const float* wn1 = (const float*)d_in[1];
  const float* bn1 = (const float*)d_in[2];
  const float* wn2 = (const float*)d_in[3];
  const float* bn2 = (const float*)d_in[4];
  const float* wf1 = (const float*)d_in[5];
  const float* bf1 = (const float*)d_in[6];
  const float* wf2 = (const float*)d_in[7];
  const float* bf2 = (const float*)d_in[8];
  const float* we1 = (const float*)d_in[9];   // [Z+R, HID]
  const float* be1 = (const float*)d_in[10];
  const float* we2 = (const float*)d_in[11];
  const float* be2 = (const float*)d_in[12];
  const float* wi1 = (const float*)d_in[13];
  const float* bi1 = (const float*)d_in[14];
  const float* wi2 = (const float*)d_in[15];
  const float* bi2 = (const float*)d_in[16];
  const float* wg1 = (const float*)d_in[17];
  const float* bg1 = (const float*)d_in[18];
  const float* wg2 = (const float*)d_in[19];
  const float* bg2 = (const float*)d_in[20];
  const float* gwx = (const float*)d_in[21];
  const float* gwh = (const float*)d_in[22];
  const float* gbx = (const float*)d_in[23];
  const float* gbh = (const float*)d_in[24];
  const float* we1_up  = we1;                  // rows [0, Z)
  const float* we1_low = we1 + (size_t)ZDIM * HIDD;  // rows [Z, Z+R)

  // Workspace layout (floats)
  float* ws = (float*)d_ws;
  float* h_nodes = ws;                               // 64*128*256 = 2097152
  float* P       = h_nodes + (size_t)MAXN * BATCH * RDIM;   // 64*128*128
  float* Q       = P + (size_t)MAXN * BATCH * HIDD;  // 128*128
  float* hc      = Q + (size_t)BATCH * HIDD;         // 64*128
  float* erow    = hc + (size_t)MAXN * BATCH;        // 64*128
  float* anc     = erow + (size_t)MAXN * BATCH;      // 128*256
  float* snk     = anc + (size_t)BATCH * RDIM;       // 128*256
  float* XZ      = snk + (size_t)BATCH * RDIM;       // 128*768
  float* HZ      = XZ + (size_t)BATCH * G3;          // 128*768
  float* S       = HZ + (size_t)BATCH * G3;          // 128*64
  float* Hg      = S + (size_t)BATCH * AGGD;         // 128*128
  float* lamh    = Hg + (size_t)BATCH * ZDIM;        // 128*128
  float* ih      = lamh + (size_t)BATCH * HIDD;      // 128*128
  float* t1      = P;  // reuse P region in epilogue (8192*128)

  // Output layout (floats)
  float* outLam = (float*)d_out;                         // [B]
  float* outF   = outLam + BATCH;                        // [B,64,F]
  float* outE   = outF + (size_t)BATCH * MAXN * FDIM;    // [B,64,64]
  int*   outN   = (int*)(outE + (size_t)BATCH * MAXN * MAXN);  // [B] int32

  const dim3 w32(32);

  // ---- Prologue -----------------------------------------------------------
  {
    const size_t nz = (size_t)(MAXN - 1) * BATCH * RDIM;
    k_zero<<<dim3((unsigned)((nz + 255) / 256)), dim3(256), 0, stream>>>(
        h_nodes, P, hc, S, bg1, outE);
  }
  // lambdas
  k_gemm<<<dim3(8, 8), w32, 0, stream>>>(x, wn1, bn1, lamh, ZDIM, HIDD, 3);
  k_lambda<<<dim3(1), dim3(BATCH), 0, stream>>>(lamh, wn2, bn2, outLam, outN);
  // init module -> h_nodes[0]
  k_gemm<<<dim3(8, 8), w32, 0, stream>>>(x, wi1, bi1, ih, ZDIM, HIDD, 3);
  k_gemm<<<dim3(8, 16), w32, 0, stream>>>(ih, wi2, bi2, h_nodes, HIDD, RDIM, 3);
  // P[0] = h0 @ we1_low ; S += relu(h0@wg1+bg1) - relu(bg1)
  k_gemm<<<dim3(8, 8), w32, 0, stream>>>(h_nodes, we1_low, nullptr, P, RDIM, HIDD, 0);
  k_gemm_supd<<<dim3(8, 4), w32, 0, stream>>>(h_nodes, wg1, bg1, S, RDIM, AGGD);
  // Q = x @ we1_up   (H_graph starts as x)
  k_gemm<<<dim3(8, 8), w32, 0, stream>>>(x, we1_up, nullptr, Q, ZDIM, HIDD, 0);

  // ---- Scan over nodes ----------------------------------------------------
  for (int i = 1; i < MAXN; ++i) {
    float* h_i = h_nodes + (size_t)i * BATCH * RDIM;
    k_edge<<<dim3(MAXN), dim3(BATCH), 0, stream>>>(Q, P, be1, we2, be2, erow, hc, outE, i);
    k_aggr<<<dim3(BATCH), dim3(RDIM), 0, stream>>>(erow, hc, h_nodes, anc, snk);
    k_gemm<<<dim3(8, 48), w32, 0, stream>>>(snk, gwx, gbx, XZ, RDIM, G3, 1);
    k_gemm<<<dim3(8, 48), w32, 0, stream>>>(anc, gwh, gbh, HZ, RDIM, G3, 1);
    k_gru<<<dim3(BATCH), dim3(RDIM), 0, stream>>>(XZ, HZ, anc, h_i);
    // incremental caches for the next steps
    k_gemm<<<dim3(8, 8), w32, 0, stream>>>(h_i, we1_low, nullptr,
                                           P + (size_t)i * BATCH * HIDD, RDIM, HIDD, 0);
    k_gemm_supd<<<dim3(8, 4), w32, 0, stream>>>(h_i, wg1, bg1, S, RDIM, AGGD);
    // H_graph = relu(S@wg2+bg2) ; Q = H_graph @ we1_up
    k_gemm<<<dim3(8, 8), w32, 0, stream>>>(S, wg2, bg2, Hg, AGGD, ZDIM, 3);
    k_gemm<<<dim3(8, 8), w32, 0, stream>>>(Hg, we1_up, nullptr, Q, ZDIM, HIDD, 0);
  }

  // ---- Epilogue: feature module + softmax ---------------------------------
  k_gemm<<<dim3(512, 8), w32, 0, stream>>>(h_nodes, wf1, bf1, t1, RDIM, HIDD, 3);
  k_gemm_fout<<<dim3(512, 8), w32, 0, stream>>>(t1, wf2, bf2, outF, HIDD, FDIM);
  k_softmax<<<dim3(MAXN * BATCH), dim3(FDIM), 0, stream>>>(outF);
}